// NussinovFold_1331439861801
// MI455X (gfx1250) — compile-verified
//
#include <hip/hip_runtime.h>
#include <stdint.h>

// Nussinov max-score DP for MI455X (gfx1250).
// One workgroup per batch element; entire DP triangle resident in LDS
// (CDNA5: 320KB/WGP). 255 barrier-synchronized wavefront steps.
// sp anti-diagonal gathered one step ahead via global_load_async_to_lds_b32
// (ASYNCcnt) into a double buffer, waited with s_wait_asynccnt.

#define NN 256
#define MIN_LOOP 3
#define NEGINF (-1000000000.0f)

// Packed lower-triangle storage: row i holds columns 0..(NN-i).
// base(i) = sum_{k<i} (NN+1-k) = i*(NN+1) - i*(i-1)/2
__device__ __forceinline__ int baseof(int i) {
    return i * (NN + 1) - ((i * (i - 1)) >> 1);
}

// LDS byte offset of a shared-memory pointer: flat LDS addresses keep the
// offset in the low 32 bits (CDNA5 ISA 10.2 aperture mapping).
__device__ __forceinline__ uint32_t lds_off32(const void* p) {
    return (uint32_t)(uintptr_t)p;
}

__global__ __launch_bounds__(256, 1)
void nussinov_kernel(const float* __restrict__ sp,   // (B, NN, NN)
                     const float* __restrict__ su,   // (B, NN)
                     float* __restrict__ out)        // (B,)
{
    // Triangle: 33,153 floats = 132,612 B. Total static LDS ~143.9 KB.
    __shared__ float tri[NN * (NN + 1) - ((NN * (NN - 1)) >> 1) + 1];
    __shared__ float su_s[NN + 1];
    __shared__ float part[8 * 256];      // replica-wave partial maxima
    __shared__ float spdiag[2][256];     // double-buffered sp anti-diagonal

    const int b   = blockIdx.x;
    const int tid = threadIdx.x;
    const float* spb = sp + (size_t)b * NN * NN;
    const float* sub = su + (size_t)b * NN;

    // init: column 1 of E = su (rows 0..NN-1); row NN never read at col>=1.
    if (tid < NN) {
        float s = sub[tid];
        su_s[tid] = s;
        tri[baseof(tid) + 1] = s;
    }
    __syncthreads();

    for (int l = 2; l <= NN; ++l) {
        const int n_i = NN - l + 1;      // valid i: 0..n_i-1

        // --- async prefetch of NEXT step's sp anti-diagonal ---------------
        // step l+1 (when l+1-1 > MIN_LOOP) needs sp[i][i+l], i in [0, NN-l-1]
        if (l >= MIN_LOOP + 1 && l < NN && tid < NN - l) {
            uint64_t ga = (uint64_t)(uintptr_t)(spb + (size_t)tid * NN + (size_t)(tid + l));
            uint32_t la = lds_off32(&spdiag[(l + 1) & 1][tid]);
            asm volatile("global_load_async_to_lds_b32 %0, %1, off"
                         :: "v"(la), "v"(ga)
                         : "memory");
        }

        // --- m-split configuration: P replica waves per i-group -----------
        const int G = (n_i + 31) >> 5;   // 32-lane groups needed to cover i
        int Gp, gsh;
        if      (G <= 1) { Gp = 1; gsh = 0; }
        else if (G <= 2) { Gp = 2; gsh = 1; }
        else if (G <= 4) { Gp = 4; gsh = 2; }
        else             { Gp = 8; gsh = 3; }
        const int P    = 8 >> gsh;       // replicas splitting the m range
        const int w    = tid >> 5;
        const int lane = tid & 31;
        const int g    = w & (Gp - 1);   // i-group
        const int r    = w >> gsh;       // replica index 0..P-1
        const int i    = (g << 5) + lane;

        // --- bifurcation partial max: max_m tri(i,m) + tri(i+m, l-m) ------
        float pm = NEGINF;
        if (i < n_i) {
            const int m0 = 1 + r;
            int a1 = baseof(i) + m0;                 // tri(i, m)
            int a2 = baseof(i + m0) + (l - m0);      // tri(i+m, l-m)
            // a2(m+P) - a2(m) = P*(NN - i) - P*(P-1)/2 - P*m
            const int c0 = P * (NN - i) - ((P * (P - 1)) >> 1);
            #pragma unroll 2
            for (int m = m0; m <= l - 1; m += P) {
                pm = fmaxf(pm, tri[a1] + tri[a2]);
                a1 += P;
                a2 += c0 - P * m;
            }
        }
        part[(r << 8) + i] = pm;

        // own-wave async gather must land before the cross-wave barrier
        asm volatile("s_wait_asynccnt 0x0" ::: "memory");
        __syncthreads();

        // --- combine: unpaired / pair / bifurcation, write tri(i, l) ------
        if (tid < n_i) {
            float v = part[tid];
            for (int r2 = 1; r2 < P; ++r2)
                v = fmaxf(v, part[(r2 << 8) + tid]);

            const float e_sh = tri[baseof(tid + 1) + (l - 1)]; // E[i+1][l-1]
            const float e_c1 = tri[baseof(tid) + (l - 1)];     // E[i][l-1]
            v = fmaxf(v, su_s[tid] + e_sh);                    // i unpaired
            v = fmaxf(v, su_s[tid + l - 1] + e_c1);            // j unpaired
            if (l - 1 > MIN_LOOP) {                            // i pairs j
                v = fmaxf(v, spdiag[l & 1][tid] + tri[baseof(tid + 1) + (l - 2)]);
            }
            tri[baseof(tid) + l] = v;
        }
        __syncthreads();
    }

    if (tid == 0) out[b] = tri[NN];      // tri(0, NN) = E[b, 0, N]
}

extern "C" void kernel_launch(void* const* d_in, const int* in_sizes, int n_in,
                              void* d_out, int out_size, void* d_ws, size_t ws_size,
                              hipStream_t stream) {
    const float* sp = (const float*)d_in[0];   // score_paired  (B, N, N) f32
    const float* su = (const float*)d_in[1];   // score_unpaired (B, N)   f32
    float* out = (float*)d_out;                // (B,) f32
    const int B = in_sizes[1] / NN;            // = 4
    nussinov_kernel<<<dim3(B), dim3(256), 0, stream>>>(sp, su, out);
}